// FFMModel_52553219834448
// MI455X (gfx1250) — compile-verified
//
#include <hip/hip_runtime.h>
#include <math.h>

typedef __attribute__((ext_vector_type(2))) float v2f;
typedef __attribute__((ext_vector_type(8))) float v8f;

#define NUM_FIELDS 10
#define VOCAB      100000
#define EMBED      8

// One wave32 handles 16 batch elements.
// WMMA f32 16x16x4 diagonal trick:
//   A[m,k] = left-vector stream of batch m, B[k,n] = right-vector stream of batch n
//   => D[b,b] accumulates sum_{i<j} dot(V[i,j,x_bj], V[j,i,x_bi])
// Per ISA 7.12.2, lane L holds, for both A and B fragments, the two K values
//   k0 = 4t + (L<16 ? 0 : 2), k0+1   for matrix row/col index (L % 16).
// k maps to (pair p, dim d) via k = 8p + d, so each lane's two K values are a
// contiguous float2 inside one 32B embedding vector -> direct global_load_b64 gathers.
__global__ __launch_bounds__(256) void ffm_wmma_kernel(
    const int*   __restrict__ x,
    const float* __restrict__ bias_p,
    const float* __restrict__ Lw,
    const float* __restrict__ V,
    float*       __restrict__ out,
    int batch)
{
    const int lane  = threadIdx.x & 31;
    const int wave  = threadIdx.x >> 5;
    const int tile  = blockIdx.x * (blockDim.x >> 5) + wave;   // 16 batches / wave
    int b = tile * 16 + (lane & 15);
    // clamp (never taken for batch=16384) so EXEC stays all-ones for WMMA
    const int bc = (b < batch) ? b : (batch - 1);
    const int koff = (lane >= 16) ? 2 : 0;

    // ---- load this batch element's index row (10 ints) ----
    int xr[NUM_FIELDS];
#pragma unroll
    for (int f = 0; f < NUM_FIELDS; ++f)
        xr[f] = x[bc * NUM_FIELDS + f];

    // ---- first-order term: sum_f L[f, x[b,f]] ----
    float first = 0.0f;
#pragma unroll
    for (int f = 0; f < NUM_FIELDS; ++f)
        first += Lw[f * VOCAB + xr[f]];

    // ---- second-order term via WMMA accumulation ----
    v8f acc = {};
#pragma unroll
    for (int i = 0; i < NUM_FIELDS; ++i) {
#pragma unroll
        for (int j = i + 1; j < NUM_FIELDS; ++j) {
            const float* pa = V + ((size_t)((i * NUM_FIELDS + j) * VOCAB + xr[j])) * EMBED;
            const float* pb = V + ((size_t)((j * NUM_FIELDS + i) * VOCAB + xr[i])) * EMBED;
            // step 1: dims [0..3] of this pair
            v2f a0 = *(const v2f*)(pa + koff);
            v2f b0 = *(const v2f*)(pb + koff);
            acc = __builtin_amdgcn_wmma_f32_16x16x4_f32(
                false, a0, false, b0, (short)0, acc, false, false);
            // step 2: dims [4..7] of this pair
            v2f a1 = *(const v2f*)(pa + 4 + koff);
            v2f b1 = *(const v2f*)(pb + 4 + koff);
            acc = __builtin_amdgcn_wmma_f32_16x16x4_f32(
                false, a1, false, b1, (short)0, acc, false, false);
        }
    }

    // ---- diagonal extraction ----
    // batch m in 0..7  : acc element m   on lane m        (lanes 0..7)
    // batch m in 8..15 : acc element m-8 on lane m+16     (lanes 24..31)
    const int r = lane & 7;
    float diag = acc[0];
    if (r == 1) diag = acc[1];
    if (r == 2) diag = acc[2];
    if (r == 3) diag = acc[3];
    if (r == 4) diag = acc[4];
    if (r == 5) diag = acc[5];
    if (r == 6) diag = acc[6];
    if (r == 7) diag = acc[7];

    const bool writer = (lane < 8) || (lane >= 24);
    if (writer && b < batch) {
        float z = bias_p[0] + first + diag;
        out[b] = 1.0f / (1.0f + __expf(-z));
    }
}

extern "C" void kernel_launch(void* const* d_in, const int* in_sizes, int n_in,
                              void* d_out, int out_size, void* d_ws, size_t ws_size,
                              hipStream_t stream) {
    // setup_inputs order: x (int), bias (f32 scalar), L (f32), V (f32)
    const int*   x    = (const int*)d_in[0];
    const float* bias = (const float*)d_in[1];
    const float* Lw   = (const float*)d_in[2];
    const float* V    = (const float*)d_in[3];
    float*       out  = (float*)d_out;

    const int batch = in_sizes[0] / NUM_FIELDS;        // 16384
    const int waves = (batch + 15) / 16;               // 16 batches per wave
    const int blocks = (waves + 7) / 8;                // 8 waves (256 thr) per block

    ffm_wmma_kernel<<<blocks, 256, 0, stream>>>(x, bias, Lw, V, out, batch);
}